// AngularMinPooling_74646531604818
// MI455X (gfx1250) — compile-verified
//
#include <hip/hip_runtime.h>
#include <hip/hip_bf16.h>
#include <stdint.h>

// AngularMinPooling on MI455X (gfx1250, wave32).
//
// Memory-bound streaming kernel: 1.024 GB read + 16 MB write => ~45us floor
// at 23.3 TB/s. Strategy: per-wave double-buffered async global->LDS DMA
// (GLOBAL_LOAD_ASYNC_TO_LDS_B128 / ASYNCcnt) to pipeline the 2KB-per-vertex
// stream, conflict-free striped LDS reads, VALU f32 sum-of-squares reduction,
// lane permutes for the 8-way argmin, LDS gather for the output.

#define WAVES_PER_BLOCK 8
#define BLOCK_SIZE (WAVES_PER_BLOCK * 32)
#define N_ROT 8
#define FEAT 64
#define VERT_FLOATS (N_ROT * FEAT)      /* 512 floats  */
#define VERT_BYTES (VERT_FLOATS * 4)    /* 2048 bytes  */

// Issue the async DMA for one vertex: 4 x b128, 32 lanes x 16B each = 2048B.
// The 24-bit instruction offset is added to BOTH the global and the LDS
// address (CDNA5 ISA), so a single (lds_off, gaddr) pair
// covers the whole vertex, leaving LDS a verbatim image of the vertex block.
__device__ __forceinline__ void issue_async_vertex(uint32_t lds_off,
                                                   const float* gbase,
                                                   int v, int lane) {
  uint64_t gaddr = (uint64_t)(uintptr_t)gbase +
                   (uint64_t)v * VERT_BYTES + (uint64_t)(lane * 16);
  asm volatile(
      "global_load_async_to_lds_b128 %0, %1, off\n\t"
      "global_load_async_to_lds_b128 %0, %1, off offset:512\n\t"
      "global_load_async_to_lds_b128 %0, %1, off offset:1024\n\t"
      "global_load_async_to_lds_b128 %0, %1, off offset:1536"
      :
      : "v"(lds_off), "v"(gaddr)
      : "memory");
}

// Reduce one staged vertex (512 floats, verbatim image in LDS) and write its
// 8 outputs.
//
// Bank-conflict-free read mapping: instruction i, lane L reads dwords
// [4L + 128i, 4L + 128i + 4). Per instruction, lanes 0-15 touch banks 0-63
// exactly once and lanes 16-31 once more: 128 dwords / 64 banks = minimum
// 2 cycles, zero conflicts. Chunk (L, i) belongs to rotation
// r = 2i + (L >= 16), since floor((4L + 128i)/64) = 2i + (L >= 16).
__device__ __forceinline__ void process_vertex(const float* __restrict__ buf,
                                               float* __restrict__ out,
                                               int v, int lane) {
  float s[4];
#pragma unroll
  for (int i = 0; i < 4; ++i) {
    float4 x = *reinterpret_cast<const float4*>(buf + 4 * lane + 128 * i);
    float t = x.x * x.x;
    t = fmaf(x.y, x.y, t);
    t = fmaf(x.z, x.z, t);
    t = fmaf(x.w, x.w, t);
    s[i] = t;
  }

  // Fold each partial across its 16-lane half: xor masks 1,2,4,8 leave bit 4
  // fixed, so lanes 0-15 (even rotations) and 16-31 (odd rotations) reduce
  // independently. Afterwards every lane of a half holds the full sum.
#pragma unroll
  for (int m = 1; m <= 8; m <<= 1) {
#pragma unroll
    for (int i = 0; i < 4; ++i) {
      s[i] += __shfl_xor(s[i], m, 32);
    }
  }

  // Broadcast: rotation 2i lives in lanes 0-15 (take lane 0), rotation 2i+1
  // in lanes 16-31 (take lane 16). All lanes end with all 8 squared norms.
  float norm[N_ROT];
#pragma unroll
  for (int i = 0; i < 4; ++i) {
    norm[2 * i + 0] = __shfl(s[i], 0, 32);
    norm[2 * i + 1] = __shfl(s[i], 16, 32);
  }

  // Wave-uniform argmin in rotation order (sqrt is monotonic, so comparing
  // squared norms preserves argmin; strict '<' keeps the first minimum like
  // jnp.argmin).
  float best = norm[0];
  int idx = 0;
#pragma unroll
  for (int r = 1; r < N_ROT; ++r) {
    if (norm[r] < best) {
      best = norm[r];
      idx = r;
    }
  }

  // out[v, r] = inputs[v, r, idx]  (idx indexes the FEATURE axis, 0..7)
  if (lane < N_ROT) {
    out[(size_t)v * N_ROT + lane] = buf[lane * FEAT + idx];
  }
}

__global__ void __launch_bounds__(BLOCK_SIZE)
angular_min_pool_kernel(const float* __restrict__ in, float* __restrict__ out,
                        int n_vertices) {
  __shared__ float lds[WAVES_PER_BLOCK][2][VERT_FLOATS];  // 32 KB / block

  const int lane = threadIdx.x & 31;
  const int wave = threadIdx.x >> 5;
  const int nwaves = gridDim.x * WAVES_PER_BLOCK;

  int v = blockIdx.x * WAVES_PER_BLOCK + wave;
  if (v >= n_vertices) return;

  // LDS byte offsets for this lane's 16B slot in each buffer (low 32 bits of
  // the generic address of a __shared__ object == LDS offset).
  const uint32_t lds0 =
      (uint32_t)(uintptr_t)(&lds[wave][0][0]) + (uint32_t)(lane * 16);
  const uint32_t lds1 =
      (uint32_t)(uintptr_t)(&lds[wave][1][0]) + (uint32_t)(lane * 16);

  // Software pipeline, depth 2: buffer `cur` in flight -> wait -> process,
  // while buffer `cur^1` is being filled for the next vertex.
  issue_async_vertex(lds0, in, v, lane);
  int cur = 0;
  for (;;) {
    const int vn = v + nwaves;
    if (vn < n_vertices) {
      // WAR guard: previous iteration's LDS reads of this buffer must have
      // completed before the DMA may overwrite it.
      asm volatile("s_wait_dscnt 0x0" ::: "memory");
      issue_async_vertex(cur ? lds0 : lds1, in, vn, lane);
      // 4 newer ops outstanding => asynccnt<=4 means the previous batch of 4
      // (buffer `cur`) has landed (async loads complete in order).
      asm volatile("s_wait_asynccnt 0x4" ::: "memory");
    } else {
      asm volatile("s_wait_asynccnt 0x0" ::: "memory");
    }

    process_vertex(&lds[wave][cur][0], out, v, lane);

    if (vn >= n_vertices) break;
    v = vn;
    cur ^= 1;
  }
}

extern "C" void kernel_launch(void* const* d_in, const int* in_sizes, int n_in,
                              void* d_out, int out_size, void* d_ws,
                              size_t ws_size, hipStream_t stream) {
  const float* in = (const float*)d_in[0];
  float* out = (float*)d_out;

  const int n_vertices = in_sizes[0] / VERT_FLOATS;  // 500000 for reference

  // Enough waves to saturate HBM while keeping a multi-vertex grid-stride
  // loop per wave so the 2-deep async pipeline pays off.
  int blocks = 3072;
  const int max_blocks = (n_vertices + WAVES_PER_BLOCK - 1) / WAVES_PER_BLOCK;
  if (blocks > max_blocks) blocks = max_blocks;
  if (blocks < 1) blocks = 1;

  hipLaunchKernelGGL(angular_min_pool_kernel, dim3(blocks), dim3(BLOCK_SIZE),
                     0, stream, in, out, n_vertices);
}